// TransConv_50981261804371
// MI455X (gfx1250) — compile-verified
//
#include <hip/hip_runtime.h>
#include <hip/hip_bf16.h>

// Problem constants (match reference)
#define NTOT   65536
#define CCH    64
#define BG     64
#define NPER   1024
#define DEG    16
#define ETOT   (NTOT * DEG)

typedef float v2f __attribute__((ext_vector_type(2)));
typedef float v8f __attribute__((ext_vector_type(8)));

// CDNA5 fp32 WMMA: D(16x16) += A(16x4) x B(4x16), fp32 throughout.
// Fragment mapping (ISA 7.12.2):
//   A: lane L holds A[m=L&15, k = v + 2*(L>>4)]  (v = vgpr index 0..1)
//   B: lane L holds B[k = v + 2*(L>>4), n=L&15]
//   C/D: lane L, vgpr r holds C[m = r + 8*(L>>4), n = L&15]
__device__ __forceinline__ v8f wmma_f32x4(v2f a, v2f b, v8f c) {
  return __builtin_amdgcn_wmma_f32_16x16x4_f32(
      /*neg_a=*/false, a, /*neg_b=*/false, b,
      /*c_mod=*/(short)0, c, /*reuse_a=*/false, /*reuse_b=*/false);
}

// ---------------------------------------------------------------------------
// 0) zero degree array + norm accumulators (contiguous in workspace)
// ---------------------------------------------------------------------------
__global__ void k_init(float* __restrict__ p, int n) {
  for (int i = blockIdx.x * blockDim.x + threadIdx.x; i < n;
       i += gridDim.x * blockDim.x)
    p[i] = 0.0f;
}

// ---------------------------------------------------------------------------
// 1a) q GEMM fused with sum-of-squares of q.
//     grid = N/128; block = 256 (8 waves). Wave = one 16-node tile,
//     full 64 output features (4 col tiles), K-loop of 16 wmma steps.
// ---------------------------------------------------------------------------
__global__ __launch_bounds__(256) void k_gemm_q(
    const float* __restrict__ X, const float* __restrict__ W,
    const float* __restrict__ bias, float* __restrict__ Y,
    float* __restrict__ norms) {
  const int lane = threadIdx.x & 31;
  const int wave = threadIdx.x >> 5;
  const int nl   = lane & 15;   // tile-local row (A) / col (B, C/D)
  const int gh   = lane >> 4;   // lane-group: K-half for A/B, +8 rows for C/D
  const int row0 = (blockIdx.x * 8 + wave) * 16;

  v8f acc[4];
#pragma unroll
  for (int t = 0; t < 4; ++t) {
    const float b0 = bias[t * 16 + nl];
#pragma unroll
    for (int r = 0; r < 8; ++r) acc[t][r] = b0;
  }

  const float* xrow = X + (size_t)(row0 + nl) * CCH;
#pragma unroll 4
  for (int kc = 0; kc < 16; ++kc) {
    const int kk = kc * 4 + gh * 2;
    const v2f a = *(const v2f*)(xrow + kk);
#pragma unroll
    for (int t = 0; t < 4; ++t) {
      const v2f b = *(const v2f*)(W + (size_t)(t * 16 + nl) * CCH + kk);
      acc[t] = wmma_f32x4(a, b, acc[t]);
    }
  }

  float sq = 0.f;
#pragma unroll
  for (int t = 0; t < 4; ++t)
#pragma unroll
    for (int r = 0; r < 8; ++r) {
      const float y = acc[t][r];
      Y[(size_t)(row0 + r + 8 * gh) * CCH + t * 16 + nl] = y;
      sq += y * y;
    }

  // block-level reduction of sum(q^2), then one atomic per block
#pragma unroll
  for (int o = 16; o > 0; o >>= 1) sq += __shfl_down(sq, o);
  __shared__ float sh[8];
  if (lane == 0) sh[wave] = sq;
  __syncthreads();
  if (threadIdx.x == 0) {
    float a = 0.f;
    for (int i = 0; i < 8; ++i) a += sh[i];
    atomicAdd(&norms[0], a);
  }
}

// ---------------------------------------------------------------------------
// 1b) k and v GEMMs fused (one read of source_input, shared A fragments),
//     plus sum-of-squares of k. 8 WMMAs per K-step.
// ---------------------------------------------------------------------------
__global__ __launch_bounds__(256) void k_gemm_kv(
    const float* __restrict__ X,
    const float* __restrict__ Wk, const float* __restrict__ bk,
    const float* __restrict__ Wv, const float* __restrict__ bv,
    float* __restrict__ ko, float* __restrict__ vo,
    float* __restrict__ norms) {
  const int lane = threadIdx.x & 31;
  const int wave = threadIdx.x >> 5;
  const int nl   = lane & 15;
  const int gh   = lane >> 4;
  const int row0 = (blockIdx.x * 8 + wave) * 16;

  v8f ak[4], av[4];
#pragma unroll
  for (int t = 0; t < 4; ++t) {
    const float b0 = bk[t * 16 + nl];
    const float b1 = bv[t * 16 + nl];
#pragma unroll
    for (int r = 0; r < 8; ++r) { ak[t][r] = b0; av[t][r] = b1; }
  }

  const float* xrow = X + (size_t)(row0 + nl) * CCH;
#pragma unroll 2
  for (int kc = 0; kc < 16; ++kc) {
    const int kk = kc * 4 + gh * 2;
    const v2f a = *(const v2f*)(xrow + kk);
#pragma unroll
    for (int t = 0; t < 4; ++t) {
      const v2f bkf = *(const v2f*)(Wk + (size_t)(t * 16 + nl) * CCH + kk);
      const v2f bvf = *(const v2f*)(Wv + (size_t)(t * 16 + nl) * CCH + kk);
      ak[t] = wmma_f32x4(a, bkf, ak[t]);
      av[t] = wmma_f32x4(a, bvf, av[t]);
    }
  }

  float sk = 0.f;
#pragma unroll
  for (int t = 0; t < 4; ++t)
#pragma unroll
    for (int r = 0; r < 8; ++r) {
      const size_t idx = (size_t)(row0 + r + 8 * gh) * CCH + t * 16 + nl;
      const float y = ak[t][r];
      ko[idx] = y;
      vo[idx] = av[t][r];
      sk += y * y;
    }

#pragma unroll
  for (int o = 16; o > 0; o >>= 1) sk += __shfl_down(sk, o);
  __shared__ float sh[8];
  if (lane == 0) sh[wave] = sk;
  __syncthreads();
  if (threadIdx.x == 0) {
    float a = 0.f;
    for (int i = 0; i < 8; ++i) a += sh[i];
    atomicAdd(&norms[1], a);
  }
}

// ---------------------------------------------------------------------------
// 3) per-graph KV = K^T V (64x64), plus k_sum / v_sum. One block per graph.
//    8 waves: wave w owns rows ci = w>>1 (16 k-channels) and two 16-wide
//    v-channel tiles; shares the A fragment between its two accumulators.
//    K dimension = 1024 nodes -> 256 wmma steps per tile.
// ---------------------------------------------------------------------------
__global__ __launch_bounds__(256) void k_graph_stats(
    const float* __restrict__ k, const float* __restrict__ v,
    float* __restrict__ kv, float* __restrict__ k_sum,
    float* __restrict__ v_sum) {
  const int g    = blockIdx.x;
  const int lane = threadIdx.x & 31;
  const int wave = threadIdx.x >> 5;
  const int nl   = lane & 15;
  const int gh   = lane >> 4;

  const float* kg = k + (size_t)g * NPER * CCH;
  const float* vg = v + (size_t)g * NPER * CCH;

  const int ci  = wave >> 1;          // k-channel tile (rows of KV)
  const int cj0 = (wave & 1) * 2;     // first of two v-channel tiles

  v8f acc0 = {0.f, 0.f, 0.f, 0.f, 0.f, 0.f, 0.f, 0.f};
  v8f acc1 = acc0;

  const int ai  = ci * 16 + nl;       // this lane's k-channel (A row m)
  const int bj0 = cj0 * 16 + nl;      // v-channel for tile 0 (B col n)
  const int bj1 = bj0 + 16;           // v-channel for tile 1

  for (int kc = 0; kc < NPER / 4; ++kc) {
    const int n0 = kc * 4 + gh * 2;   // node index for fragment element 0
    v2f a, b0, b1;
    a[0]  = kg[(size_t)n0 * CCH + ai];
    a[1]  = kg[(size_t)(n0 + 1) * CCH + ai];
    b0[0] = vg[(size_t)n0 * CCH + bj0];
    b0[1] = vg[(size_t)(n0 + 1) * CCH + bj0];
    b1[0] = vg[(size_t)n0 * CCH + bj1];
    b1[1] = vg[(size_t)(n0 + 1) * CCH + bj1];
    acc0 = wmma_f32x4(a, b0, acc0);
    acc1 = wmma_f32x4(a, b1, acc1);
  }

  float* kvg = kv + (size_t)g * CCH * CCH;
#pragma unroll
  for (int r = 0; r < 8; ++r) {
    const int m = ci * 16 + r + 8 * gh;
    kvg[(size_t)m * CCH + bj0] = acc0[r];
    kvg[(size_t)m * CCH + bj1] = acc1[r];
  }

  // k_sum / v_sum: 4 partials per channel across 256 threads
  __shared__ float shv[4][CCH];
  __shared__ float shk[4][CCH];
  const int c    = threadIdx.x & 63;
  const int part = threadIdx.x >> 6;
  float sv = 0.f, sk = 0.f;
  for (int n = part; n < NPER; n += 4) {
    sv += vg[(size_t)n * CCH + c];
    sk += kg[(size_t)n * CCH + c];
  }
  shv[part][c] = sv;
  shk[part][c] = sk;
  __syncthreads();
  if (threadIdx.x < CCH) {
    v_sum[g * CCH + c] = shv[0][c] + shv[1][c] + shv[2][c] + shv[3][c];
    k_sum[g * CCH + c] = shk[0][c] + shk[1][c] + shk[2][c] + shk[3][c];
  }
}

// ---------------------------------------------------------------------------
// 4) attention = (s * (q @ KV_raw) + v_sum) / (s * (q . k_sum_raw) + n_nodes)
//    where s = 1/(||q|| * ||k||). Wave = one 16-node tile.
// ---------------------------------------------------------------------------
__global__ __launch_bounds__(256) void k_attention(
    const float* __restrict__ q, const float* __restrict__ kv,
    const float* __restrict__ k_sum, const float* __restrict__ v_sum,
    const float* __restrict__ norms, const int* __restrict__ n_nodes,
    float* __restrict__ out) {
  const int lane = threadIdx.x & 31;
  const int wave = threadIdx.x >> 5;
  const int nl   = lane & 15;
  const int gh   = lane >> 4;
  const int row0 = (blockIdx.x * 8 + wave) * 16;
  const int g    = row0 / NPER;

  const float s  = rsqrtf(norms[0]) * rsqrtf(norms[1]);
  const float nn = (float)n_nodes[g];
  const float* kvg = kv + (size_t)g * CCH * CCH;

  v8f acc[4];
#pragma unroll
  for (int t = 0; t < 4; ++t)
#pragma unroll
    for (int r = 0; r < 8; ++r) acc[t][r] = 0.f;

  const float* qrow = q + (size_t)(row0 + nl) * CCH;
#pragma unroll 4
  for (int kc = 0; kc < 16; ++kc) {
    const int kk = kc * 4 + gh * 2;
    const v2f a = *(const v2f*)(qrow + kk);
#pragma unroll
    for (int t = 0; t < 4; ++t) {
      v2f b;
      b[0] = kvg[(size_t)kk * CCH + t * 16 + nl];
      b[1] = kvg[(size_t)(kk + 1) * CCH + t * 16 + nl];
      acc[t] = wmma_f32x4(a, b, acc[t]);
    }
  }

  // denominator dot product: lane L computes it for tile-row (L&15)
  float dsum = 0.f;
  {
    const float* ks = k_sum + g * CCH;
#pragma unroll 8
    for (int c = 0; c < CCH; ++c) dsum += qrow[c] * ks[c];
  }

#pragma unroll
  for (int r = 0; r < 8; ++r) {
    const float dm    = __shfl(dsum, r + 8 * gh);
    const float rdet  = 1.0f / (s * dm + nn);
    const int   m     = row0 + r + 8 * gh;
#pragma unroll
    for (int t = 0; t < 4; ++t) {
      const float vs = v_sum[g * CCH + t * 16 + nl];
      out[(size_t)m * CCH + t * 16 + nl] = (s * acc[t][r] + vs) * rdet;
    }
  }
}

// ---------------------------------------------------------------------------
// 5) degree histogram over col indices
// ---------------------------------------------------------------------------
__global__ __launch_bounds__(256) void k_degree(const int* __restrict__ col,
                                                float* __restrict__ deg) {
  for (int e = blockIdx.x * blockDim.x + threadIdx.x; e < ETOT;
       e += gridDim.x * blockDim.x)
    atomicAdd(&deg[col[e]], 1.0f);
}

// ---------------------------------------------------------------------------
// 6) GCN term accumulated into out. One wave per node (row[e] = e/DEG by
//    construction). Lane l covers channels l and l+32.
// ---------------------------------------------------------------------------
__global__ __launch_bounds__(256) void k_gcn(const int* __restrict__ col,
                                             const float* __restrict__ deg,
                                             const float* __restrict__ v,
                                             float* __restrict__ out) {
  const int lane = threadIdx.x & 31;
  const int wave = threadIdx.x >> 5;
  const int node = blockIdx.x * 8 + wave;

  const float dn = deg[node];
  int   cj   = 0;
  float valj = 0.f;
  if (lane < DEG) {
    cj = col[node * DEG + lane];
    const float dc = deg[cj];
    if (dn > 0.f && dc > 0.f) valj = rsqrtf(dc) * rsqrtf(dn);
  }

  float a0 = 0.f, a1 = 0.f;
#pragma unroll
  for (int j = 0; j < DEG; ++j) {
    const int   c = __shfl(cj, j);
    const float w = __shfl(valj, j);
    a0 += w * v[(size_t)c * CCH + lane];
    a1 += w * v[(size_t)c * CCH + 32 + lane];
  }
  out[(size_t)node * CCH + lane]      += a0;
  out[(size_t)node * CCH + 32 + lane] += a1;
}

// ---------------------------------------------------------------------------
// launch
// ---------------------------------------------------------------------------
extern "C" void kernel_launch(void* const* d_in, const int* in_sizes, int n_in,
                              void* d_out, int out_size, void* d_ws,
                              size_t ws_size, hipStream_t stream) {
  const float* query_input  = (const float*)d_in[0];
  const float* source_input = (const float*)d_in[1];
  const int*   n_nodes      = (const int*)d_in[2];
  const int*   edge_index   = (const int*)d_in[3];  // [2, E]; col at +ETOT
  const float* Wq_w = (const float*)d_in[4];
  const float* Wq_b = (const float*)d_in[5];
  const float* Wk_w = (const float*)d_in[6];
  const float* Wk_b = (const float*)d_in[7];
  const float* Wv_w = (const float*)d_in[8];
  const float* Wv_b = (const float*)d_in[9];
  float* out = (float*)d_out;

  const size_t NC = (size_t)NTOT * CCH;
  float* q     = (float*)d_ws;
  float* k     = q + NC;
  float* v     = k + NC;
  float* kv    = v + NC;                      // B*64*64
  float* v_sum = kv + (size_t)BG * CCH * CCH; // B*64
  float* k_sum = v_sum + (size_t)BG * CCH;    // B*64
  float* deg   = k_sum + (size_t)BG * CCH;    // N
  float* norms = deg + NTOT;                  // 2 (contiguous with deg)

  const int* edge_col = edge_index + ETOT;

  // 0) zero deg + norms (contiguous region of NTOT+2 floats)
  k_init<<<256, 256, 0, stream>>>(deg, NTOT + 2);

  // 1) GEMMs via V_WMMA_F32_16X16X4_F32, with fused norm reductions
  k_gemm_q<<<NTOT / 128, 256, 0, stream>>>(query_input, Wq_w, Wq_b, q, norms);
  k_gemm_kv<<<NTOT / 128, 256, 0, stream>>>(source_input, Wk_w, Wk_b, Wv_w,
                                            Wv_b, k, v, norms);

  // 3) per-graph K^T V + k_sum + v_sum (one WGP block per graph)
  k_graph_stats<<<BG, 256, 0, stream>>>(k, v, kv, k_sum, v_sum);

  // 4) attention output (writes all of out)
  k_attention<<<NTOT / 128, 256, 0, stream>>>(q, kv, k_sum, v_sum, norms,
                                              n_nodes, out);

  // 5) degree histogram
  k_degree<<<2048, 256, 0, stream>>>(edge_col, deg);

  // 6) GCN gather-accumulate into out
  k_gcn<<<NTOT / 8, 256, 0, stream>>>(edge_col, deg, v, out);
}